// EMAQuantizer_46007689675065
// MI455X (gfx1250) — compile-verified
//
#include <hip/hip_runtime.h>

// ---------------------------------------------------------------------------
// VQ-VAE quantizer forward for MI455X (gfx1250, wave32, WMMA).
// distances GEMM fused with argmin using V_WMMA_F32_16X16X4_F32 (exact fp32).
// ---------------------------------------------------------------------------

typedef __attribute__((ext_vector_type(2))) float v2f;
typedef __attribute__((ext_vector_type(4))) float v4f;
typedef __attribute__((ext_vector_type(8))) float v8f;

#define NUM_EMB      1024
#define CDIM         64
#define NPTS         65536          // B*D*H*W = 2*32*32*32
#define SPATIAL      32768          // D*H*W
#define CH_STRIDE    32768          // channel stride in elements (channel-first)
#define BATCH_STRIDE (64 * 32768)

// d_out layout (floats), outputs concatenated in reference return order:
#define OUT_Q_OFF    0              // quantized_st  [2,64,32,32,32] = 4194304
#define OUT_LOSS_OFF 4194304        // loss scalar
#define OUT_IDX_OFF  4194305        // encoding_indices [2,32,32,32] = 65536
#define OUT_ESUM_OFF 4259841        // encodings_sum zeros(256)
#define OUT_EMB_OFF  4260097        // embedding passthrough = 65536

// workspace layout (floats)
#define WS_HNORM     0              // 1024 half-norms 0.5*||e_k||^2
#define WS_LOSSACC   1024           // 1 accumulator

// ---- half-norm precompute + loss accumulator reset -------------------------
__global__ void vq_hnorm_kernel(const float* __restrict__ emb,
                                float* __restrict__ ws) {
    int k = blockIdx.x * blockDim.x + threadIdx.x;
    if (k == 0) ws[WS_LOSSACC] = 0.0f;
    if (k < NUM_EMB) {
        const v4f* row = (const v4f*)(emb + (size_t)k * CDIM);
        float s = 0.0f;
#pragma unroll
        for (int i = 0; i < CDIM / 4; ++i) {
            v4f t = row[i];
            s += t.x * t.x + t.y * t.y + t.z * t.z + t.w * t.w;
        }
        ws[WS_HNORM + k] = 0.5f * s;
    }
}

// ---- fused distance-GEMM + argmin + gather + loss ---------------------------
// One wave handles 16 points; block = 4 waves (128 threads); 1024 blocks.
__global__ void __launch_bounds__(128)
vq_main_kernel(const float* __restrict__ x, const float* __restrict__ emb,
               const float* __restrict__ ws, float* __restrict__ lossacc,
               float* __restrict__ out) {
    __shared__ int sIdx[4][16];

    const int lane  = threadIdx.x & 31;
    const int wv    = threadIdx.x >> 5;
    const int n0    = (blockIdx.x * 4 + wv) * 16;   // first point of this wave
    const int b     = n0 >> 15;                     // / SPATIAL
    const int sp0   = n0 & (SPATIAL - 1);
    const int row   = lane & 15;                    // matrix row (M) / col (N)
    const int chalf = (lane >> 4) * 2;              // K-subchunk select per half-wave

    // ---- load A: 16 rows x 64 channels of flat_input (strided, fp32) ------
    // WMMA A(16x4) layout: lanes 0-15 hold K={c,c+1}, lanes 16-31 hold K={c+2,c+3}
    const float* xbase = x + (size_t)b * BATCH_STRIDE + sp0 + row;
    v2f   a[16];
    float ssq = 0.0f;                               // per-lane partial ||x_row||^2
#pragma unroll
    for (int ci = 0; ci < 16; ++ci) {
        float a0 = xbase[(size_t)(4 * ci + chalf)     * CH_STRIDE];
        float a1 = xbase[(size_t)(4 * ci + chalf + 1) * CH_STRIDE];
        a[ci].x = a0;
        a[ci].y = a1;
        ssq += a0 * a0 + a1 * a1;
    }

    float bestv[8];
    int   besti[8];
#pragma unroll
    for (int v = 0; v < 8; ++v) { bestv[v] = -3.4e38f; besti[v] = 0; }

    // ---- sweep 64 tiles of 16 codes; 16 chained fp32 WMMAs per tile --------
    for (int kt = 0; kt < NUM_EMB / 16; ++kt) {
        const float* erow = emb + (size_t)(kt * 16 + row) * CDIM + chalf;
        v2f bm[16];
#pragma unroll
        for (int ci = 0; ci < 16; ++ci)
            bm[ci] = *(const v2f*)(erow + 4 * ci);

        v8f acc = {0.f, 0.f, 0.f, 0.f, 0.f, 0.f, 0.f, 0.f};
#pragma unroll
        for (int ci = 0; ci < 16; ++ci)
            acc = __builtin_amdgcn_wmma_f32_16x16x4_f32(
                false, a[ci], false, bm[ci], (short)0, acc, false, false);

        const float hn   = ws[WS_HNORM + kt * 16 + row];
        const int   kidx = kt * 16 + row;
#pragma unroll
        for (int v = 0; v < 8; ++v) {
            float s = acc[v] - hn;                  // x.e_k - 0.5||e_k||^2
            if (s > bestv[v]) { bestv[v] = s; besti[v] = kidx; }
        }
    }

    // ---- argmax butterfly over the 16 columns in each half-wave ------------
#pragma unroll
    for (int v = 0; v < 8; ++v) {
#pragma unroll
        for (int off = 8; off >= 1; off >>= 1) {
            float ov = __shfl_xor(bestv[v], off, 32);
            int   oi = __shfl_xor(besti[v], off, 32);
            if (ov > bestv[v] || (ov == bestv[v] && oi < besti[v])) {
                bestv[v] = ov; besti[v] = oi;
            }
        }
    }
    // full ||x_row||^2: combine the two channel halves (lane L with L^16)
    float rssq = ssq + __shfl_xor(ssq, 16, 32);

    // ---- loss partials + encoding-index output + LDS broadcast of indices --
    // loss per row n: sum_c (q - x)^2 = ||x||^2 - 2 * best_score  (see analysis)
    float  lossPart = 0.0f;
    float* encOut   = out + OUT_IDX_OFF + n0;
#pragma unroll
    for (int v = 0; v < 8; ++v) {
        if (lane == v) {                      // rows 0..7 (lanes 0-15 group)
            lossPart += rssq - 2.0f * bestv[v];
            encOut[v] = (float)besti[v];
            sIdx[wv][v] = besti[v];
        }
        if (lane == 16 + v) {                 // rows 8..15 (lanes 16-31 group)
            encOut[8 + v] = (float)besti[v];
            sIdx[wv][8 + v] = besti[v];
        }
        if (lane == 24 + v) {                 // lane 24+v holds rssq of row 8+v
            lossPart += rssq - 2.0f * bestv[v];
        }
    }
#pragma unroll
    for (int off = 16; off >= 1; off >>= 1)
        lossPart += __shfl_xor(lossPart, off, 32);
    if (lane == 0) atomicAdd(lossacc, lossPart);

    __syncthreads();

    // ---- write quantized_st = embedding[idx], channel-first, coalesced -----
    const int    myIdx = sIdx[wv][row];
    const float* qrow  = emb + (size_t)myIdx * CDIM + (lane >> 4) * 32;
    float*       qout  = out + OUT_Q_OFF + (size_t)b * BATCH_STRIDE + sp0 + row;
#pragma unroll
    for (int it = 0; it < 8; ++it) {
        v4f qv = *(const v4f*)(qrow + it * 4);
        int ch = (lane >> 4) * 32 + it * 4;
        qout[(size_t)(ch + 0) * CH_STRIDE] = qv.x;
        qout[(size_t)(ch + 1) * CH_STRIDE] = qv.y;
        qout[(size_t)(ch + 2) * CH_STRIDE] = qv.z;
        qout[(size_t)(ch + 3) * CH_STRIDE] = qv.w;
    }
}

// ---- finalize scalar loss ---------------------------------------------------
__global__ void vq_loss_final_kernel(const float* __restrict__ ws,
                                     float* __restrict__ out) {
    if (threadIdx.x == 0 && blockIdx.x == 0)
        out[OUT_LOSS_OFF] = 0.25f * ws[WS_LOSSACC] / 4194304.0f; // / (N*C)
}

// ---- encodings_sum zeros + embedding passthrough ----------------------------
__global__ void vq_tail_kernel(const float* __restrict__ emb,
                               float* __restrict__ out) {
    int i = blockIdx.x * blockDim.x + threadIdx.x;
    if (i < 256) out[OUT_ESUM_OFF + i] = 0.0f;
    if (i < NUM_EMB * CDIM) out[OUT_EMB_OFF + i] = emb[i];
}

extern "C" void kernel_launch(void* const* d_in, const int* in_sizes, int n_in,
                              void* d_out, int out_size, void* d_ws, size_t ws_size,
                              hipStream_t stream) {
    const float* x   = (const float*)d_in[0];   // [2,64,32,32,32] fp32
    const float* emb = (const float*)d_in[1];   // [1024,64] fp32
    float*       out = (float*)d_out;
    float*       ws  = (float*)d_ws;

    vq_hnorm_kernel<<<dim3(NUM_EMB / 256), dim3(256), 0, stream>>>(emb, ws);
    vq_main_kernel<<<dim3(NPTS / 64), dim3(128), 0, stream>>>(
        x, emb, ws, ws + WS_LOSSACC, out);
    vq_loss_final_kernel<<<dim3(1), dim3(32), 0, stream>>>(ws, out);
    vq_tail_kernel<<<dim3((NUM_EMB * CDIM) / 256), dim3(256), 0, stream>>>(emb, out);
}